// GibbsSampler_40381282517204
// MI455X (gfx1250) — compile-verified
//
#include <hip/hip_runtime.h>
#include <stdint.h>

// Gibbs sampler, restructured:
//   Phase 0: per-step threefry fold_in keys (1024 hashes).
//   Phase 1: all Gumbel thresholds C[b,t], affine terms A[b,t], mask M, init V,
//            written TRANSPOSED [t][b] for coalesced Phase-2 access. Parallel.
//   Phase 2: 256 independent sequential chains (1 lane per row), double-buffered
//            CDNA5 async global->LDS staging (ASYNCcnt) to hide latency; inner
//            loop is pure LDS+VALU; outputs batched into b128 stores.

#define LDIM 1024
#define BDIM 256
#define SCALE_C 2988.0f
#define EDGE_C  300.0f
#define LAM_C   510.3f

// ---------------- threefry2x32 (exact JAX algorithm) ----------------
__device__ __forceinline__ uint32_t rotl32(uint32_t v, int n) {
  return (v << n) | (v >> (32 - n));
}

__device__ __forceinline__ void tf2x32(uint32_t k0, uint32_t k1,
                                       uint32_t x0, uint32_t x1,
                                       uint32_t& y0, uint32_t& y1) {
  const uint32_t ks2 = k0 ^ k1 ^ 0x1BD11BDAu;
  x0 += k0; x1 += k1;
#define TF_R4(a,b,c,d) \
  x0 += x1; x1 = rotl32(x1,(a)); x1 ^= x0; \
  x0 += x1; x1 = rotl32(x1,(b)); x1 ^= x0; \
  x0 += x1; x1 = rotl32(x1,(c)); x1 ^= x0; \
  x0 += x1; x1 = rotl32(x1,(d)); x1 ^= x0;
  TF_R4(13,15,26, 6)  x0 += k1;  x1 += ks2 + 1u;
  TF_R4(17,29,16,24)  x0 += ks2; x1 += k0  + 2u;
  TF_R4(13,15,26, 6)  x0 += k0;  x1 += k1  + 3u;
  TF_R4(17,29,16,24)  x0 += k1;  x1 += ks2 + 4u;
  TF_R4(13,15,26, 6)  x0 += ks2; x1 += k0  + 5u;
#undef TF_R4
  y0 = x0; y1 = x1;
}

// JAX gumbel: u = bitcast(bits>>9 | 0x3f800000) - 1; u = max(tiny, u + tiny);
// g = -log(-log(u))
__device__ __forceinline__ float gumbel_f(uint32_t bits) {
  const float tiny = 1.17549435e-38f;
  float u = __uint_as_float((bits >> 9) | 0x3f800000u) - 1.0f;
  u = fmaxf(tiny, u + tiny);
  return -__logf(-__logf(u));
}

// ---------------- Phase 0: per-step keys = fold_in(key(42), t) ----------------
__global__ void gibbs_keys(uint32_t* __restrict__ keys) {
  int t = blockIdx.x * blockDim.x + threadIdx.x;
  if (t < LDIM) {
    uint32_t k0, k1;
    tf2x32(0u, 42u, 0u, (uint32_t)t, k0, k1);  // base key = (0, 42)
    keys[2 * t]     = k0;
    keys[2 * t + 1] = k1;
  }
}

// ---------------- Phase 1: thresholds + affine terms (transposed) ----------------
__global__ __launch_bounds__(256) void gibbs_precompute(
    const float* __restrict__ mask, const float* __restrict__ s,
    const int* __restrict__ init, const uint32_t* __restrict__ keys,
    float* __restrict__ A_T, float* __restrict__ C_T,
    float* __restrict__ M_T, float* __restrict__ V_T) {
  const int t = blockIdx.x;    // one timestep per block
  const int b = threadIdx.x;   // one batch row per lane
  const uint32_t k0 = keys[2 * t], k1 = keys[2 * t + 1];

  // Flattened (B,2) draw index f = 2b+j; counters split (f, f+256), n=512.
  const uint32_t f0 = 2u * (uint32_t)b;
  uint32_t g0b, g1b, y0, y1;
  if (b < 128) {                                   // f < 256 -> take y0
    tf2x32(k0, k1, f0,        f0 + 256u, y0, y1); g0b = y0;
    tf2x32(k0, k1, f0 + 1u,   f0 + 257u, y0, y1); g1b = y0;
  } else {                                         // f >= 256 -> take y1
    tf2x32(k0, k1, f0 - 256u, f0,        y0, y1); g0b = y1;
    tf2x32(k0, k1, f0 - 255u, f0 + 1u,   y0, y1); g1b = y1;
  }

  const int gi = b * LDIM + t;   // original [B,L] layout
  const int oi = t * BDIM + b;   // transposed [L,B] layout
  const float mk = mask[gi];
  C_T[oi] = gumbel_f(g0b) - gumbel_f(g1b);
  A_T[oi] = mk * (s[gi] * SCALE_C - LAM_C);
  M_T[oi] = mk;
  V_T[oi] = (float)init[gi] * mk;
}

// ---------------- CDNA5 async global->LDS copy + wait ----------------
// Inline asm (assembler-verified last round). Flat LDS address low 32 bits ==
// wave-relative LDS byte offset (ISA 10.2 aperture mapping).
__device__ __forceinline__ void async_cp_b128(const void* g, void* l) {
  asm volatile("global_load_async_to_lds_b128 %0, %1, off"
               :: "v"((uint32_t)(uintptr_t)l), "v"(g) : "memory");
}

#if __has_builtin(__builtin_amdgcn_s_wait_asynccnt)
#define WAIT_ASYNC(n) __builtin_amdgcn_s_wait_asynccnt(n)
#else
#define WAIT_ASYNC(n) asm volatile("s_wait_asynccnt %0" :: "i"(n) : "memory")
#endif

// ---------------- Phase 2: 256 sequential chains, pipelined LDS staging ----------------
#define CH     8                 // timesteps per staged chunk
#define NCHUNK (LDIM / CH)       // 128 chunks
#define CHF    (CH * BDIM)       // 2048 floats per array per chunk

__global__ __launch_bounds__(256) void gibbs_scan(
    const float* __restrict__ A_T, const float* __restrict__ C_T,
    const float* __restrict__ M_T, const float* __restrict__ V_T,
    int* __restrict__ out) {
  __shared__ float buf[2][4][CHF];   // double-buffered A/C/M/V chunks: 64 KB
  const int b = threadIdx.x;

  auto stage = [&](int c, int pb) {
    const float* gsrc[4] = {A_T + c * CHF, C_T + c * CHF,
                            M_T + c * CHF, V_T + c * CHF};
#pragma unroll
    for (int a0 = 0; a0 < 4; ++a0) {
#pragma unroll
      for (int i = 0; i < 2; ++i) {           // 2048 floats = 512 f4; 2 per thread
        const int idx4 = i * 256 + b;
        async_cp_b128(gsrc[a0] + idx4 * 4, &buf[pb][a0][idx4 * 4]);
      }
    }
  };

  stage(0, 0);
  float vprev = 0.0f;                          // updated m[t-1]*mask[t-1]
  for (int c = 0; c < NCHUNK; ++c) {
    const int pb = c & 1;
    if (c + 1 < NCHUNK) {
      stage(c + 1, pb ^ 1);                    // prefetch next chunk
      WAIT_ASYNC(8);                           // 8 outstanding = next chunk only
    } else {
      WAIT_ASYNC(0);
    }
    // Boundary vnext (t = c*CH+CH-1 needs V of next chunk; next chunk's LDS
    // copy is not guaranteed landed yet, so take it from global, hoisted here
    // so it overlaps the LDS work below).
    const float vbound = (c + 1 < NCHUNK) ? V_T[(c + 1) * CHF + b] : 0.0f;
    __syncthreads();                           // publish this chunk's LDS data

    int zarr[CH];
#pragma unroll
    for (int tt = 0; tt < CH; ++tt) {
      const float a  = buf[pb][0][tt * BDIM + b];
      const float cc = buf[pb][1][tt * BDIM + b];
      const float mk = buf[pb][2][tt * BDIM + b];
      // m[t+1] still holds its ORIGINAL init value during a left->right sweep
      const float vnext = (tt < CH - 1) ? buf[pb][3][(tt + 1) * BDIM + b]
                                        : vbound;
      const float delta = fmaf(EDGE_C * mk, vprev + vnext, a);
      const int z = (delta > cc) ? 1 : 0;      // argmax ties -> 0, strict >
      zarr[tt] = z;
      vprev = (float)z * mk;
    }
    // Lane b owns 8 consecutive ints of out: two 16B stores, 16B-aligned.
    int4* orow = (int4*)(out + b * LDIM + c * CH);
    orow[0] = make_int4(zarr[0], zarr[1], zarr[2], zarr[3]);
    orow[1] = make_int4(zarr[4], zarr[5], zarr[6], zarr[7]);
    __syncthreads();                           // protect buffer reuse (c+2 overwrite)
  }
}

// ---------------- host launch ----------------
extern "C" void kernel_launch(void* const* d_in, const int* in_sizes, int n_in,
                              void* d_out, int out_size, void* d_ws, size_t ws_size,
                              hipStream_t stream) {
  (void)in_sizes; (void)n_in; (void)out_size; (void)ws_size;
  const float* mask = (const float*)d_in[0];   // attention_mask [B,L] f32
  const float* s    = (const float*)d_in[1];   // s [B,L] f32
  const int*   init = (const int*)d_in[2];     // init [B,L] i32
  int* out = (int*)d_out;                      // [B,L] i32

  // Scratch layout: 4 x [L,B] f32 (4 MB) + 1024 key pairs (8 KB)
  float* A_T = (float*)d_ws;
  float* C_T = A_T + BDIM * LDIM;
  float* M_T = C_T + BDIM * LDIM;
  float* V_T = M_T + BDIM * LDIM;
  uint32_t* keys = (uint32_t*)(V_T + BDIM * LDIM);

  gibbs_keys<<<(LDIM + 255) / 256, 256, 0, stream>>>(keys);
  gibbs_precompute<<<LDIM, BDIM, 0, stream>>>(mask, s, init, keys,
                                              A_T, C_T, M_T, V_T);
  gibbs_scan<<<1, BDIM, 0, stream>>>(A_T, C_T, M_T, V_T, out);
}